// DownUpLayer_352187318293
// MI455X (gfx1250) — compile-verified
//
#include <hip/hip_runtime.h>

typedef float v2f __attribute__((ext_vector_type(2)));
typedef float v8f __attribute__((ext_vector_type(8)));

constexpr int kH = 256;   // hidden dim
constexpr int kB = 64;    // bottleneck dim
constexpr int kWaves = 2; // waves (16-node tiles) per block

// ---------------------------------------------------------------- zero agg
__global__ void zero_f4_kernel(float* __restrict__ p, size_t n4) {
  size_t i = (size_t)blockIdx.x * blockDim.x + threadIdx.x;
  if (i < n4) reinterpret_cast<float4*>(p)[i] = make_float4(0.f, 0.f, 0.f, 0.f);
}

// --------------------------------------------------- edge scatter (segment_sum)
// one thread per (edge, 16B chunk): agg[dst] += x[src]
__global__ void gin_scatter_kernel(const float* __restrict__ x,
                                   const long long* __restrict__ src,
                                   const long long* __restrict__ dst,
                                   float* __restrict__ agg,
                                   long long E) {
  long long tid = (long long)blockIdx.x * blockDim.x + threadIdx.x;
  if (tid >= E * (kH / 4)) return;
  long long e = tid >> 6;               // kH/4 == 64 chunks per edge
  int c4 = (int)(tid & 63);
  long long s = src[e];
  long long d = dst[e];
  const float4 v = *reinterpret_cast<const float4*>(x + (size_t)s * kH + c4 * 4);
  float* ap = agg + (size_t)d * kH + c4 * 4;
  atomicAdd(ap + 0, v.x);
  atomicAdd(ap + 1, v.y);
  atomicAdd(ap + 2, v.z);
  atomicAdd(ap + 3, v.w);
}

// --------------------------------------------- fused GIN MLP + residual + LN
// per 16-node tile (one wave32 per tile):
//   h = (1+eps)*x + agg                                [16 x 256]
//   t = relu(LN_64(h @ W1))                            [16 x 64]
//   y = LN_256(relu(t @ W2 + x + dir_emb))             [16 x 256]
__launch_bounds__(kWaves * 32)
__global__ void gin_fused_kernel(const float* __restrict__ x,
                                 const float* __restrict__ agg,
                                 const float* __restrict__ eps_p,
                                 const float* __restrict__ W1,   // [256][64]
                                 const float* __restrict__ lng,  // [64]
                                 const float* __restrict__ lnb,  // [64]
                                 const float* __restrict__ W2,   // [64][256]
                                 const float* __restrict__ lnHg, // [256]
                                 const float* __restrict__ lnHb, // [256]
                                 const float* __restrict__ demb, // [256]
                                 float* __restrict__ out,
                                 int N) {
  // wave-private LDS slices (no cross-wave sharing -> no barriers needed)
  __shared__ float lds_hT[kWaves][16][kH + 4];  // h staging, reused for epilogue
  __shared__ float lds_t1[kWaves][16][kB + 4];  // bottleneck activations
  __shared__ float lds_m[kWaves][16];
  __shared__ float lds_rs[kWaves][16];

  const int lane = threadIdx.x & 31;
  const int wave = threadIdx.x >> 5;
  const int tile = blockIdx.x * kWaves + wave;
  const int node0 = tile * 16;
  if (node0 >= N) return;  // wave-uniform exit; EXEC stays all-1s for WMMA

  float (*hT)[kH + 4] = lds_hT[wave];
  float (*t1)[kB + 4] = lds_t1[wave];
  float* rm = lds_m[wave];
  float* rrs = lds_rs[wave];

  const float eps1 = 1.0f + eps_p[0];

  // ---- build h into LDS (coalesced global reads; data is L2-resident) ----
  for (int idx = lane; idx < 16 * kH; idx += 32) {
    int r = idx >> 8;
    int c = idx & (kH - 1);
    int node = node0 + r;
    float v = 0.f;
    if (node < N) {
      size_t o = (size_t)node * kH + c;
      v = eps1 * x[o] + agg[o];
    }
    hT[r][c] = v;
  }

  // f32 WMMA 16x16x4 fragment coordinates (ISA 7.12.2)
  const int Mrow = lane & 15;            // A: row
  const int Ncol = lane & 15;            // B/C: column within 16-tile
  const int koff = (lane >> 4) << 1;     // K pair base: 0 or 2
  const int mbase = (lane >> 4) << 3;    // C/D: M base 0 or 8

  // ---- GEMM1: t1 = h(16x256) @ W1(256x64), 4 N-tiles, K-loop of 4 ----
  v8f acc0 = {}, acc1 = {}, acc2 = {}, acc3 = {};
  for (int k = 0; k < kH; k += 4) {
    v2f a;
    a.x = hT[Mrow][k + koff];
    a.y = hT[Mrow][k + koff + 1];
    const float* w0 = W1 + (size_t)(k + koff) * kB;
    const float* w1 = W1 + (size_t)(k + koff + 1) * kB;
    v2f b0, b1, b2, b3;
    b0.x = w0[Ncol];      b0.y = w1[Ncol];
    b1.x = w0[16 + Ncol]; b1.y = w1[16 + Ncol];
    b2.x = w0[32 + Ncol]; b2.y = w1[32 + Ncol];
    b3.x = w0[48 + Ncol]; b3.y = w1[48 + Ncol];
    acc0 = __builtin_amdgcn_wmma_f32_16x16x4_f32(false, a, false, b0, (short)0, acc0, false, false);
    acc1 = __builtin_amdgcn_wmma_f32_16x16x4_f32(false, a, false, b1, (short)0, acc1, false, false);
    acc2 = __builtin_amdgcn_wmma_f32_16x16x4_f32(false, a, false, b2, (short)0, acc2, false, false);
    acc3 = __builtin_amdgcn_wmma_f32_16x16x4_f32(false, a, false, b3, (short)0, acc3, false, false);
  }
  // scatter C fragments to LDS t1[16][64]
  for (int vv = 0; vv < 8; ++vv) {
    t1[mbase + vv][0 * 16 + Ncol] = acc0[vv];
    t1[mbase + vv][1 * 16 + Ncol] = acc1[vv];
    t1[mbase + vv][2 * 16 + Ncol] = acc2[vv];
    t1[mbase + vv][3 * 16 + Ncol] = acc3[vv];
  }

  // ---- LN over 64 + ReLU, in place (2 lanes per row) ----
  {
    int r = lane >> 1;
    int cb = (lane & 1) * 32;
    float s = 0.f;
    for (int c = 0; c < 32; ++c) s += t1[r][cb + c];
    s += __shfl_xor(s, 1, 32);
    float mean = s * (1.0f / 64.0f);
    float vs = 0.f;
    for (int c = 0; c < 32; ++c) { float d = t1[r][cb + c] - mean; vs += d * d; }
    vs += __shfl_xor(vs, 1, 32);
    float rstd = rsqrtf(vs * (1.0f / 64.0f) + 1e-5f);
    for (int c = 0; c < 32; ++c) {
      int cc = cb + c;
      float val = (t1[r][cc] - mean) * rstd * lng[cc] + lnb[cc];
      t1[r][cc] = fmaxf(val, 0.f);
    }
  }

  // ---- GEMM2: t1(16x64) @ W2(64x256); 4 groups of 4 N-tiles ----
  for (int g = 0; g < 4; ++g) {
    const int n0 = g * 64;
    v8f c0 = {}, c1 = {}, c2 = {}, c3 = {};
    for (int k = 0; k < kB; k += 4) {
      v2f a;
      a.x = t1[Mrow][k + koff];
      a.y = t1[Mrow][k + koff + 1];
      const float* w0 = W2 + (size_t)(k + koff) * kH + n0;
      const float* w1 = W2 + (size_t)(k + koff + 1) * kH + n0;
      v2f b0, b1, b2, b3;
      b0.x = w0[Ncol];      b0.y = w1[Ncol];
      b1.x = w0[16 + Ncol]; b1.y = w1[16 + Ncol];
      b2.x = w0[32 + Ncol]; b2.y = w1[32 + Ncol];
      b3.x = w0[48 + Ncol]; b3.y = w1[48 + Ncol];
      c0 = __builtin_amdgcn_wmma_f32_16x16x4_f32(false, a, false, b0, (short)0, c0, false, false);
      c1 = __builtin_amdgcn_wmma_f32_16x16x4_f32(false, a, false, b1, (short)0, c1, false, false);
      c2 = __builtin_amdgcn_wmma_f32_16x16x4_f32(false, a, false, b2, (short)0, c2, false, false);
      c3 = __builtin_amdgcn_wmma_f32_16x16x4_f32(false, a, false, b3, (short)0, c3, false, false);
    }
    // epilogue stage: relu(h2 + x + dir_emb) into hT columns [n0, n0+64)
    for (int vv = 0; vv < 8; ++vv) {
      int node = node0 + mbase + vv;
      const float* xr = x + (size_t)(node < N ? node : N - 1) * kH;
      int col;
      col = n0 + Ncol;
      hT[mbase + vv][col] = fmaxf(c0[vv] + xr[col] + demb[col], 0.f);
      col = n0 + 16 + Ncol;
      hT[mbase + vv][col] = fmaxf(c1[vv] + xr[col] + demb[col], 0.f);
      col = n0 + 32 + Ncol;
      hT[mbase + vv][col] = fmaxf(c2[vv] + xr[col] + demb[col], 0.f);
      col = n0 + 48 + Ncol;
      hT[mbase + vv][col] = fmaxf(c3[vv] + xr[col] + demb[col], 0.f);
    }
  }

  // ---- final LN over 256 (2 lanes per row for stats) ----
  {
    int r = lane >> 1;
    int cb = (lane & 1) * 128;
    float s = 0.f;
    for (int c = 0; c < 128; ++c) s += hT[r][cb + c];
    s += __shfl_xor(s, 1, 32);
    float mean = s * (1.0f / 256.0f);
    float vs = 0.f;
    for (int c = 0; c < 128; ++c) { float d = hT[r][cb + c] - mean; vs += d * d; }
    vs += __shfl_xor(vs, 1, 32);
    if ((lane & 1) == 0) {
      rm[r] = mean;
      rrs[r] = rsqrtf(vs * (1.0f / 256.0f) + 1e-5f);
    }
  }
  // coalesced store
  for (int r = 0; r < 16; ++r) {
    int node = node0 + r;
    if (node >= N) break;  // wave-uniform
    float mean = rm[r], rstd = rrs[r];
    for (int c = lane; c < kH; c += 32) {
      out[(size_t)node * kH + c] = (hT[r][c] - mean) * rstd * lnHg[c] + lnHb[c];
    }
  }
}

// ---------------------------------------------------------------- launcher
extern "C" void kernel_launch(void* const* d_in, const int* in_sizes, int n_in,
                              void* d_out, int out_size, void* d_ws, size_t ws_size,
                              hipStream_t stream) {
  const float* x = (const float*)d_in[0];
  const long long* ei = (const long long*)d_in[1];
  const float* eps_dn = (const float*)d_in[2];
  const float* W1_dn = (const float*)d_in[3];
  const float* lng_dn = (const float*)d_in[4];
  const float* lnb_dn = (const float*)d_in[5];
  const float* W2_dn = (const float*)d_in[6];
  const float* eps_up = (const float*)d_in[7];
  const float* W1_up = (const float*)d_in[8];
  const float* lng_up = (const float*)d_in[9];
  const float* lnb_up = (const float*)d_in[10];
  const float* W2_up = (const float*)d_in[11];
  const float* ln1_g = (const float*)d_in[12];
  const float* ln1_b = (const float*)d_in[13];
  const float* ln2_g = (const float*)d_in[14];
  const float* ln2_b = (const float*)d_in[15];
  const float* demb = (const float*)d_in[16];

  const int N = in_sizes[0] / kH;
  const long long E = (long long)(in_sizes[1] / 2);
  const long long* src = ei;
  const long long* dst = ei + E;

  float* agg = (float*)d_ws;                 // N*kH floats
  float* x1 = agg + (size_t)N * kH;          // N*kH floats
  float* outp = (float*)d_out;

  const size_t n4 = (size_t)N * kH / 4;
  const int zb = (int)((n4 + 255) / 256);
  const long long etot = E * (kH / 4);
  const int sb = (int)((etot + 255) / 256);
  const int tiles = (N + 15) / 16;
  const int fb = (tiles + kWaves - 1) / kWaves;

  // ---- direction 0: down (edges src -> dst) ----
  zero_f4_kernel<<<zb, 256, 0, stream>>>(agg, n4);
  gin_scatter_kernel<<<sb, 256, 0, stream>>>(x, src, dst, agg, E);
  gin_fused_kernel<<<fb, kWaves * 32, 0, stream>>>(
      x, agg, eps_dn, W1_dn, lng_dn, lnb_dn, W2_dn, ln1_g, ln1_b, demb, x1, N);

  // ---- direction 1: up (reversed edges dst -> src) ----
  zero_f4_kernel<<<zb, 256, 0, stream>>>(agg, n4);
  gin_scatter_kernel<<<sb, 256, 0, stream>>>(x1, dst, src, agg, E);
  gin_fused_kernel<<<fb, kWaves * 32, 0, stream>>>(
      x1, agg, eps_up, W1_up, lng_up, lnb_up, W2_up, ln2_g, ln2_b, demb + kH, outp, N);
}